// Model_5368709120573
// MI455X (gfx1250) — compile-verified
//
#include <hip/hip_runtime.h>
#include <math.h>

// ---- problem constants ----
constexpr int CB  = 32;    // batch
constexpr int CL  = 512;   // history length
constexpr int CN  = 64;    // variables
constexpr int CP  = 32;    // patches
constexpr int CTE = 10;    // time-encoding channels
constexpr int CH  = 128;   // hidden
constexpr int CH2 = 256;   // 2*hidden
constexpr int CLP = 96;    // prediction length
constexpr float CS = 1.0f / 32.0f;   // HIST / P
constexpr float LN_EPS = 1e-5f;

typedef __attribute__((ext_vector_type(2))) float v2f;
typedef __attribute__((ext_vector_type(8))) float v8f;

__device__ __forceinline__ v8f wmma_f32(v2f a, v2f b, v8f c) {
  // D = A(16x4) * B(4x16) + C(16x16), fp32 exact
  return __builtin_amdgcn_wmma_f32_16x16x4_f32(false, a, false, b, (short)0, c,
                                               false, false);
}

// fast sigmoid via hardware v_exp_f32 (window weights only)
__device__ __forceinline__ float fast_sigmoid(float z) {
  return 1.0f / (1.0f + __expf(-z));
}

// ---------------------------------------------------------------------------
// K0: positional-encoding table pe[P][H] (computed once, 16 KB)
// ---------------------------------------------------------------------------
__global__ void __launch_bounds__(256)
pe_kernel(float* __restrict__ pe) {
  int i = blockIdx.x * 256 + threadIdx.x;
  if (i < CP * CH) {
    const int p = i >> 7, c = i & 127;
    const int ii = c >> 1;
    const float div = expf(-(float)(2 * ii) * (9.210340371976184f / 128.0f));
    const float ang = (float)p * div;
    pe[i] = (c & 1) ? cosf(ang) : sinf(ang);
  }
}

// ---------------------------------------------------------------------------
// K1: learnable time encodings for history and prediction marks
// ---------------------------------------------------------------------------
__global__ void __launch_bounds__(256)
te_kernel(const float* __restrict__ xm, const float* __restrict__ ym,
          const float* __restrict__ sw, const float* __restrict__ sb,
          const float* __restrict__ pw, const float* __restrict__ pb,
          float* __restrict__ te_his, float* __restrict__ te_pred) {
  int i = blockIdx.x * 256 + threadIdx.x;
  const int nh = CB * CL, np = CB * CLP;
  if (i < nh) {
    float t = xm[i];
    float* o = te_his + i * CTE;
    o[0] = t * sw[0] + sb[0];
#pragma unroll
    for (int f = 0; f < CTE - 1; ++f) o[1 + f] = sinf(t * pw[f] + pb[f]);
  } else if (i < nh + np) {
    int j = i - nh;
    float t = ym[j];
    float* o = te_pred + j * CTE;
    o[0] = t * sw[0] + sb[0];
#pragma unroll
    for (int f = 0; f < CTE - 1; ++f) o[1 + f] = sinf(t * pw[f] + pb[f]);
  }
}

// ---------------------------------------------------------------------------
// K2: soft patch pooling (sigmoid window weights) + projection 11 -> 128
// one block per (b, n); 256 threads = 32 patches x 8 lanes over L
// ---------------------------------------------------------------------------
__global__ void __launch_bounds__(256)
patch_proj_kernel(const float* __restrict__ x, const float* __restrict__ xm,
                  const float* __restrict__ xmask,
                  const float* __restrict__ te_his,
                  const float* __restrict__ delta_left,
                  const float* __restrict__ rlw, const float* __restrict__ tau,
                  const float* __restrict__ proj_w,
                  const float* __restrict__ proj_b,
                  float* __restrict__ h_proj) {
  const int bx = blockIdx.x;
  const int b = bx >> 6, n = bx & 63;
  const int tid = threadIdx.x;
  const int p = tid >> 3, sub = tid & 7;

  __shared__ float sAcc[CP][12];
  __shared__ float sHn[CP][11];

  const float tl = (float)p * CS + delta_left[n * CP + p];
  const float tr = tl + expf(rlw[n * CP + p]) + 1e-6f;
  const float inv_tau = 1.0f / (log1pf(expf(tau[n])) + 1e-6f);  // 1/softplus

  float acc[12];
#pragma unroll
  for (int k = 0; k < 12; ++k) acc[k] = 0.0f;

  for (int l = sub; l < CL; l += 8) {
    const int bl = b * CL + l;
    float t = xm[bl];
    float m = xmask[bl * CN + n];
    float w = fast_sigmoid((tr - t) * inv_tau) * fast_sigmoid((t - tl) * inv_tau);
    float tw = w * m;
    acc[0] += tw;
    acc[1] += tw * x[bl * CN + n];
    const float* te = te_his + bl * CTE;
#pragma unroll
    for (int f = 0; f < CTE; ++f) acc[2 + f] += tw * te[f];
  }
  // reduce across the 8 lanes sharing a patch (contiguous lanes in a wave32)
#pragma unroll
  for (int off = 4; off >= 1; off >>= 1) {
#pragma unroll
    for (int k = 0; k < 12; ++k) acc[k] += __shfl_xor(acc[k], off);
  }
  if (sub == 0) {
#pragma unroll
    for (int k = 0; k < 12; ++k) sAcc[p][k] = acc[k];
  }
  __syncthreads();

  for (int idx = tid; idx < CP * 11; idx += 256) {
    int pp = idx / 11, f = idx % 11;
    sHn[pp][f] = sAcc[pp][1 + f] / (sAcc[pp][0] + 1e-9f);
  }
  __syncthreads();

  // h_proj[p][h] = h_avg[p][:] @ proj_w + proj_b
  for (int idx = tid; idx < CP * CH; idx += 256) {
    int pp = idx >> 7, h = idx & 127;
    float s = proj_b[h];
#pragma unroll
    for (int f = 0; f < 11; ++f) s += sHn[pp][f] * proj_w[f * CH + h];
    h_proj[(bx * CP + pp) * CH + h] = s;
  }
}

// ---------------------------------------------------------------------------
// K3: fused FFN (128->256 GELU 256->128) + residual + LayerNorm + pos-enc
// 64 rows per block, 8 waves, fp32 WMMA 16x16x4, in-place over h_proj
// ---------------------------------------------------------------------------
__global__ void __launch_bounds__(256)
ffn_ln_kernel(float* __restrict__ hbuf, const float* __restrict__ w1,
              const float* __restrict__ b1, const float* __restrict__ w2,
              const float* __restrict__ b2, const float* __restrict__ ng,
              const float* __restrict__ nb, const float* __restrict__ pe) {
  __shared__ float smem[64 * CH2];  // 64 KB: Hd (64x256), later reused as Y (64x128)
  const int tid = threadIdx.x;
  const int wave = tid >> 5, lane = tid & 31;
  const int l16 = lane & 15, hi = lane >> 4;
  const int row0 = blockIdx.x * 64;

  // ---- stage A: Hd = gelu(X @ W1 + b1) ----
  {
    const int m = wave & 3, nh = wave >> 2;
    const float* Xrow = hbuf + (row0 + m * 16 + l16) * CH;
    v8f acc[8];
#pragma unroll
    for (int t = 0; t < 8; ++t)
#pragma unroll
      for (int j = 0; j < 8; ++j) acc[t][j] = 0.0f;

    for (int k = 0; k < CH; k += 4) {
      const int kk = k + 2 * hi;
      v2f a;
      a.x = Xrow[kk];
      a.y = Xrow[kk + 1];
#pragma unroll
      for (int t = 0; t < 8; ++t) {
        const int c = nh * 128 + t * 16 + l16;
        v2f bf;
        bf.x = w1[kk * CH2 + c];
        bf.y = w1[(kk + 1) * CH2 + c];
        acc[t] = wmma_f32(a, bf, acc[t]);
      }
    }
#pragma unroll
    for (int t = 0; t < 8; ++t) {
      const int c = nh * 128 + t * 16 + l16;
#pragma unroll
      for (int j = 0; j < 8; ++j) {
        const int r = m * 16 + j + 8 * hi;
        float v = acc[t][j] + b1[c];
        v = 0.5f * v * (1.0f + erff(v * 0.70710678118654752f));  // exact GELU
        smem[r * CH2 + c] = v;
      }
    }
  }
  __syncthreads();

  // ---- stage B: Y = Hd @ W2 + b2 ----
  const int m2 = wave & 3, ng2 = wave >> 2;
  v8f acc2[4];
#pragma unroll
  for (int t = 0; t < 4; ++t)
#pragma unroll
    for (int j = 0; j < 8; ++j) acc2[t][j] = 0.0f;

  for (int k = 0; k < CH2; k += 4) {
    const int kk = k + 2 * hi;
    const float* Hrow = smem + (m2 * 16 + l16) * CH2;
    v2f a;
    a.x = Hrow[kk];
    a.y = Hrow[kk + 1];
#pragma unroll
    for (int t = 0; t < 4; ++t) {
      const int c = ng2 * 64 + t * 16 + l16;
      v2f bf;
      bf.x = w2[kk * CH + c];
      bf.y = w2[(kk + 1) * CH + c];
      acc2[t] = wmma_f32(a, bf, acc2[t]);
    }
  }
  __syncthreads();  // all Hd reads done; smem reused as Y (64x128)
#pragma unroll
  for (int t = 0; t < 4; ++t) {
    const int c = ng2 * 64 + t * 16 + l16;
#pragma unroll
    for (int j = 0; j < 8; ++j) {
      const int r = m2 * 16 + j + 8 * hi;
      smem[r * CH + c] = acc2[t][j] + b2[c];
    }
  }
  __syncthreads();

  // ---- residual + LayerNorm + positional encoding; 4 threads per row ----
  {
    const int r = tid >> 2, sub = tid & 3;
    const int grow = row0 + r;
    float* Rrow = hbuf + grow * CH;
    float vals[32];
    float sum = 0.0f, sq = 0.0f;
#pragma unroll
    for (int i = 0; i < 32; ++i) {
      const int c = sub * 32 + i;
      float v = Rrow[c] + smem[r * CH + c];
      vals[i] = v;
      sum += v;
      sq += v * v;
    }
    sum += __shfl_xor(sum, 1);
    sum += __shfl_xor(sum, 2);
    sq += __shfl_xor(sq, 1);
    sq += __shfl_xor(sq, 2);
    const float mean = sum * (1.0f / 128.0f);
    const float var = sq * (1.0f / 128.0f) - mean * mean;
    const float rstd = rsqrtf(var + LN_EPS);
    const int pp = grow & (CP - 1);  // patch index (rows ordered b,n,p)
    const float* perow = pe + pp * CH;
#pragma unroll
    for (int i = 0; i < 32; ++i) {
      const int c = sub * 32 + i;
      float v = (vals[i] - mean) * rstd * ng[c] + nb[c] + perow[c];
      Rrow[c] = v;  // in-place: h_patches overwrites h_proj
    }
  }
}

// ---------------------------------------------------------------------------
// K4: patch attention (softmax over P) + LayerNorm -> h_final (B*N x H)
// ---------------------------------------------------------------------------
__global__ void __launch_bounds__(256)
attn_kernel(const float* __restrict__ hp, const float* __restrict__ q,
            const float* __restrict__ ag, const float* __restrict__ ab,
            float* __restrict__ hfinal) {
  const int bn = blockIdx.x;
  const int n = bn & 63;
  const int tid = threadIdx.x;
  const float* base = hp + (size_t)bn * CP * CH;

  __shared__ float sS[CP];
  __shared__ float sA[8], sB[8];

  // scores: 8 lanes per patch
  {
    const int p = tid >> 3, sub = tid & 7;
    float s = 0.0f;
#pragma unroll
    for (int i = 0; i < 16; ++i) {
      const int h = sub * 16 + i;
      s += q[n * CH + h] * base[p * CH + h];
    }
    s += __shfl_xor(s, 1);
    s += __shfl_xor(s, 2);
    s += __shfl_xor(s, 4);
    if (sub == 0) sS[p] = s * 0.08838834764831845f;  // H^-0.5
  }
  __syncthreads();
  if (tid < 32) {
    float s = sS[tid];
    float mx = s;
#pragma unroll
    for (int off = 16; off >= 1; off >>= 1) mx = fmaxf(mx, __shfl_xor(mx, off));
    float e = expf(s - mx);
    float sum = e;
#pragma unroll
    for (int off = 16; off >= 1; off >>= 1) sum += __shfl_xor(sum, off);
    sS[tid] = e / sum;
  }
  __syncthreads();

  float acc = 0.0f;
  if (tid < CH) {
#pragma unroll
    for (int p = 0; p < CP; ++p) acc += sS[p] * base[p * CH + tid];
  }
  float s1 = (tid < CH) ? acc : 0.0f;
  float s2 = (tid < CH) ? acc * acc : 0.0f;
#pragma unroll
  for (int off = 16; off >= 1; off >>= 1) {
    s1 += __shfl_xor(s1, off);
    s2 += __shfl_xor(s2, off);
  }
  if ((tid & 31) == 0) {
    sA[tid >> 5] = s1;
    sB[tid >> 5] = s2;
  }
  __syncthreads();
  float t1 = 0.0f, t2 = 0.0f;
#pragma unroll
  for (int w = 0; w < 8; ++w) {
    t1 += sA[w];
    t2 += sB[w];
  }
  if (tid < CH) {
    const float mean = t1 * (1.0f / 128.0f);
    const float var = t2 * (1.0f / 128.0f) - mean * mean;
    const float rstd = rsqrtf(var + LN_EPS);
    hfinal[bn * CH + tid] = (acc - mean) * rstd * ag[tid] + ab[tid];
  }
}

// ---------------------------------------------------------------------------
// K5a: hpart = h_final (2048 x 128) @ dec_w1[0:128,:] (128 x 256), fp32 WMMA
// ---------------------------------------------------------------------------
__global__ void __launch_bounds__(256)
hpart_gemm_kernel(const float* __restrict__ hf, const float* __restrict__ dw1,
                  float* __restrict__ hpart) {
  const int tid = threadIdx.x;
  const int wave = tid >> 5, lane = tid & 31;
  const int l16 = lane & 15, hi = lane >> 4;
  const int row0 = blockIdx.x * 64;
  const int m = wave & 3, nh = wave >> 2;

  const float* Xrow = hf + (row0 + m * 16 + l16) * CH;
  v8f acc[8];
#pragma unroll
  for (int t = 0; t < 8; ++t)
#pragma unroll
    for (int j = 0; j < 8; ++j) acc[t][j] = 0.0f;

  for (int k = 0; k < CH; k += 4) {
    const int kk = k + 2 * hi;
    v2f a;
    a.x = Xrow[kk];
    a.y = Xrow[kk + 1];
#pragma unroll
    for (int t = 0; t < 8; ++t) {
      const int c = nh * 128 + t * 16 + l16;
      v2f bf;
      bf.x = dw1[kk * CH2 + c];
      bf.y = dw1[(kk + 1) * CH2 + c];
      acc[t] = wmma_f32(a, bf, acc[t]);
    }
  }
#pragma unroll
  for (int t = 0; t < 8; ++t) {
    const int c = nh * 128 + t * 16 + l16;
#pragma unroll
    for (int j = 0; j < 8; ++j) {
      const int r = row0 + m * 16 + j + 8 * hi;
      hpart[r * CH2 + c] = acc[t][j];
    }
  }
}

// ---------------------------------------------------------------------------
// K5b: te_contrib = te_pred (3072 x 10) @ dec_w1[128:138,:]
// ---------------------------------------------------------------------------
__global__ void __launch_bounds__(256)
tec_kernel(const float* __restrict__ te_pred, const float* __restrict__ dw1,
           float* __restrict__ tec) {
  int i = blockIdx.x * 256 + threadIdx.x;
  if (i < CB * CLP * CH2) {
    const int r = i >> 8, c = i & 255;
    float s = 0.0f;
#pragma unroll
    for (int f = 0; f < CTE; ++f)
      s += te_pred[r * CTE + f] * dw1[(CH + f) * CH2 + c];
    tec[i] = s;
  }
}

// ---------------------------------------------------------------------------
// K6: out[b,lp,n] = relu(hpart[b,n] + tec[b,lp] + b1) . dec_w2 + b2
// one wave per output element
// ---------------------------------------------------------------------------
__global__ void __launch_bounds__(256)
dec_kernel(const float* __restrict__ hpart, const float* __restrict__ tec,
           const float* __restrict__ b1, const float* __restrict__ w2,
           const float* __restrict__ b2, float* __restrict__ out) {
  const int wid = blockIdx.x * 8 + (threadIdx.x >> 5);
  const int lane = threadIdx.x & 31;
  const int lp = wid % CLP;
  const int bn = wid / CLP;
  const int b = bn >> 6, n = bn & 63;
  const float* hprow = hpart + (size_t)bn * CH2;
  const float* tcrow = tec + (size_t)(b * CLP + lp) * CH2;
  float s = 0.0f;
#pragma unroll
  for (int i = 0; i < 8; ++i) {
    const int c = lane + i * 32;
    float v = hprow[c] + tcrow[c] + b1[c];
    s += fmaxf(v, 0.0f) * w2[c];
  }
#pragma unroll
  for (int off = 16; off >= 1; off >>= 1) s += __shfl_xor(s, off);
  if (lane == 0) out[(b * CLP + lp) * CN + n] = s + b2[0];
}

// ---------------------------------------------------------------------------
extern "C" void kernel_launch(void* const* d_in, const int* in_sizes, int n_in,
                              void* d_out, int out_size, void* d_ws,
                              size_t ws_size, hipStream_t stream) {
  (void)in_sizes; (void)n_in; (void)out_size; (void)ws_size;
  const float* x          = (const float*)d_in[0];
  const float* x_mark     = (const float*)d_in[1];
  const float* x_mask     = (const float*)d_in[2];
  const float* y_mark     = (const float*)d_in[3];
  const float* te_scale_w = (const float*)d_in[4];
  const float* te_scale_b = (const float*)d_in[5];
  const float* te_per_w   = (const float*)d_in[6];
  const float* te_per_b   = (const float*)d_in[7];
  const float* delta_left = (const float*)d_in[8];
  const float* raw_log_w  = (const float*)d_in[9];
  const float* tau        = (const float*)d_in[10];
  const float* proj_w     = (const float*)d_in[11];
  const float* proj_b     = (const float*)d_in[12];
  const float* ffn_w1     = (const float*)d_in[13];
  const float* ffn_b1     = (const float*)d_in[14];
  const float* ffn_w2     = (const float*)d_in[15];
  const float* ffn_b2     = (const float*)d_in[16];
  const float* norm_g     = (const float*)d_in[17];
  const float* norm_b     = (const float*)d_in[18];
  const float* var_q      = (const float*)d_in[19];
  const float* agg_g      = (const float*)d_in[20];
  const float* agg_b      = (const float*)d_in[21];
  const float* dec_w1     = (const float*)d_in[22];
  const float* dec_b1     = (const float*)d_in[23];
  const float* dec_w2     = (const float*)d_in[24];
  const float* dec_b2     = (const float*)d_in[25];
  float* out = (float*)d_out;

  // workspace layout (floats), ~40.6 MB total
  float* wsf     = (float*)d_ws;
  float* te_his  = wsf;                                 // CB*CL*CTE
  float* te_pred = te_his + CB * CL * CTE;              // CB*CLP*CTE
  float* hbuf    = te_pred + CB * CLP * CTE;            // CB*CN*CP*CH (h_proj -> h_patches in place)
  float* h_final = hbuf + (size_t)CB * CN * CP * CH;    // CB*CN*CH
  float* hpart   = h_final + CB * CN * CH;              // CB*CN*CH2
  float* tec     = hpart + CB * CN * CH2;               // CB*CLP*CH2
  float* pe      = tec + CB * CLP * CH2;                // CP*CH

  pe_kernel<<<(CP * CH + 255) / 256, 256, 0, stream>>>(pe);

  te_kernel<<<(CB * CL + CB * CLP + 255) / 256, 256, 0, stream>>>(
      x_mark, y_mark, te_scale_w, te_scale_b, te_per_w, te_per_b, te_his,
      te_pred);

  patch_proj_kernel<<<CB * CN, 256, 0, stream>>>(
      x, x_mark, x_mask, te_his, delta_left, raw_log_w, tau, proj_w, proj_b,
      hbuf);

  ffn_ln_kernel<<<(CB * CN * CP) / 64, 256, 0, stream>>>(
      hbuf, ffn_w1, ffn_b1, ffn_w2, ffn_b2, norm_g, norm_b, pe);

  attn_kernel<<<CB * CN, 256, 0, stream>>>(hbuf, var_q, agg_g, agg_b, h_final);

  hpart_gemm_kernel<<<(CB * CN) / 64, 256, 0, stream>>>(h_final, dec_w1,
                                                        hpart);

  tec_kernel<<<(CB * CLP * CH2 + 255) / 256, 256, 0, stream>>>(te_pred, dec_w1,
                                                               tec);

  dec_kernel<<<(CB * CN * CLP) / 8, 256, 0, stream>>>(hpart, tec, dec_b1,
                                                      dec_w2, dec_b2, out);
}